// IterativeEncoder_49185965474341
// MI455X (gfx1250) — compile-verified
//
#include <hip/hip_runtime.h>
#include <math.h>

// IterativeEncoder for MI455X (gfx1250, wave32).
// B=8, N=2048, k=32, nkey=10 fixed by setup_inputs().
// - Heavy GEMMs (keyf@Wm1, h@Wm2): V_WMMA_F32_16X16X4_F32, A staged via
//   global_load_async_to_lds_b128 (ASYNCcnt path).
// - Score stage algebraically fused (Wk/Wd/Wc/bn3 folded on-device) and run
//   on WMMA as well (gathered 16-row tiles, K=48, N=128).

#define LRELU_S 0.2f
#define EPS_S   1e-5f
#define Bc   8
#define Nc   2048
#define BNc  (Bc * Nc)
#define KNBR 32
#define KP1  33
#define NKEY 10

typedef __attribute__((ext_vector_type(2))) float v2f;
typedef __attribute__((ext_vector_type(8))) float v8f;

__device__ __forceinline__ float lrelu(float v) { return v > 0.f ? v : LRELU_S * v; }

// LDS byte offset of a __shared__ object: generic LDS pointers carry the LDS
// offset in their low 32 bits (aperture mapping truncates addr[31:0]).
__device__ __forceinline__ unsigned lds_off(const void* p) {
  return (unsigned)(uintptr_t)p;
}

// ---------------------------------------------------------------------------
// kNN: one thread per query point, sorted candidate list in LDS (stride-128
// layout), neighbor point tile staged in LDS. Handles d<=16, k<=33.
// ---------------------------------------------------------------------------
__global__ __launch_bounds__(128) void knn_kernel(const float* __restrict__ pts,
                                                  int d, int k, int exclude_self,
                                                  float* __restrict__ out_d,
                                                  int* __restrict__ out_i) {
  __shared__ float s_d[KP1 * 128];
  __shared__ int   s_i[KP1 * 128];
  __shared__ float s_t[16 * 128];
  const int tid = threadIdx.x;
  const int q   = blockIdx.x * 128 + tid;   // [0, B*N)
  const int b   = q / Nc;
  const int i   = q % Nc;

  float qv[16];
  #pragma unroll
  for (int u = 0; u < 16; ++u) qv[u] = (u < d) ? pts[(size_t)q * d + u] : 0.f;

  for (int s = 0; s < k; ++s) { s_d[s * 128 + tid] = 3.0e38f; s_i[s * 128 + tid] = 0; }

  for (int j0 = 0; j0 < Nc; j0 += 128) {
    __syncthreads();
    for (int u = 0; u < d; ++u)
      s_t[u * 128 + tid] = pts[((size_t)b * Nc + j0 + tid) * d + u];
    __syncthreads();
    for (int jj = 0; jj < 128; ++jj) {
      const int j = j0 + jj;
      if (exclude_self && j == i) continue;
      float dist = 0.f;
      for (int u = 0; u < d; ++u) {
        float df = qv[u] - s_t[u * 128 + jj];
        dist += df * df;
      }
      if (dist < s_d[(k - 1) * 128 + tid]) {      // beats current worst
        int pos = k - 1;
        while (pos > 0 && s_d[(pos - 1) * 128 + tid] > dist) {   // strict: stable ties
          s_d[pos * 128 + tid] = s_d[(pos - 1) * 128 + tid];
          s_i[pos * 128 + tid] = s_i[(pos - 1) * 128 + tid];
          --pos;
        }
        s_d[pos * 128 + tid] = dist;
        s_i[pos * 128 + tid] = j;
      }
    }
  }
  for (int s = 0; s < k; ++s) {
    if (out_d) out_d[(size_t)q * k + s] = s_d[s * 128 + tid];
    out_i[(size_t)q * k + s] = s_i[s * 128 + tid];
  }
}

// ---------------------------------------------------------------------------
// EdgeConv1: x(B,N,3), idx(B,N,32), W1(6,16) -> x1(B,N,16)
// ---------------------------------------------------------------------------
__global__ __launch_bounds__(256) void edgeconv1_kernel(const float* __restrict__ x,
                                                        const int* __restrict__ idx,
                                                        const float* __restrict__ W1,
                                                        const float* __restrict__ b1,
                                                        float* __restrict__ x1) {
  __shared__ float sW[96];
  __shared__ float sb[16];
  const int tid = threadIdx.x;
  if (tid < 96) sW[tid] = W1[tid];
  if (tid < 16) sb[tid] = b1[tid];
  __syncthreads();
  const int q = blockIdx.x * 256 + tid;
  const int b = q / Nc;
  const float xi0 = x[(size_t)q * 3 + 0], xi1 = x[(size_t)q * 3 + 1], xi2 = x[(size_t)q * 3 + 2];
  float mx[16];
  #pragma unroll
  for (int c = 0; c < 16; ++c) mx[c] = -3.0e38f;
  for (int t = 0; t < KNBR; ++t) {
    const int j = idx[(size_t)q * KNBR + t];
    const float* xj = x + ((size_t)b * Nc + j) * 3;
    const float d0 = xj[0] - xi0, d1 = xj[1] - xi1, d2 = xj[2] - xi2;
    #pragma unroll
    for (int c = 0; c < 16; ++c) {
      float h = sb[c] + xi0 * sW[0 * 16 + c] + xi1 * sW[1 * 16 + c] + xi2 * sW[2 * 16 + c]
                      + d0 * sW[3 * 16 + c] + d1 * sW[4 * 16 + c] + d2 * sW[5 * 16 + c];
      mx[c] = fmaxf(mx[c], lrelu(h));
    }
  }
  #pragma unroll
  for (int c = 0; c < 16; ++c) x1[(size_t)q * 16 + c] = mx[c];
}

// ---------------------------------------------------------------------------
// EdgeConv2: x1(B,N,16), idx(B,N,32), W2(32,48) -> x2(B,N,48)
// ---------------------------------------------------------------------------
__global__ __launch_bounds__(256) void edgeconv2_kernel(const float* __restrict__ x1in,
                                                        const int* __restrict__ idx,
                                                        const float* __restrict__ W2,
                                                        const float* __restrict__ b2,
                                                        float* __restrict__ x2) {
  __shared__ float sW[1536];
  __shared__ float sb[48];
  const int tid = threadIdx.x;
  for (int u = tid; u < 1536; u += 256) sW[u] = W2[u];
  if (tid < 48) sb[tid] = b2[tid];
  __syncthreads();
  const int q = blockIdx.x * 256 + tid;
  const int b = q / Nc;
  float xi[16];
  #pragma unroll
  for (int u = 0; u < 16; ++u) xi[u] = x1in[(size_t)q * 16 + u];
  float base[48];
  #pragma unroll
  for (int c = 0; c < 48; ++c) {
    float acc = sb[c];
    #pragma unroll
    for (int u = 0; u < 16; ++u) acc += xi[u] * sW[u * 48 + c];
    base[c] = acc;
  }
  float mx[48];
  #pragma unroll
  for (int c = 0; c < 48; ++c) mx[c] = -3.0e38f;
  for (int t = 0; t < KNBR; ++t) {
    const int j = idx[(size_t)q * KNBR + t];
    const float* xj = x1in + ((size_t)b * Nc + j) * 16;
    float dd[16];
    #pragma unroll
    for (int u = 0; u < 16; ++u) dd[u] = xj[u] - xi[u];
    #pragma unroll
    for (int c = 0; c < 48; ++c) {
      float h = base[c];
      #pragma unroll
      for (int u = 0; u < 16; ++u) h += dd[u] * sW[(16 + u) * 48 + c];
      mx[c] = fmaxf(mx[c], lrelu(h));
    }
  }
  #pragma unroll
  for (int c = 0; c < 48; ++c) x2[(size_t)q * 48 + c] = mx[c];
}

// ---------------------------------------------------------------------------
// Prep: fold Wk/Wd/bk/bd/bc/bn3 into combined score weights, and build
// per-column (scale, shift) for the two BN+bias epilogues.
// ---------------------------------------------------------------------------
__global__ void prep_kernel(const float* __restrict__ Wk, const float* __restrict__ bk,
                            const float* __restrict__ Wd, const float* __restrict__ bd,
                            const float* __restrict__ Wc, const float* __restrict__ bc,
                            const float* __restrict__ g3, const float* __restrict__ be3,
                            const float* __restrict__ m3, const float* __restrict__ v3,
                            const float* __restrict__ bm1, const float* __restrict__ g1,
                            const float* __restrict__ be1, const float* __restrict__ m1,
                            const float* __restrict__ v1,
                            const float* __restrict__ bm2, const float* __restrict__ g2,
                            const float* __restrict__ be2, const float* __restrict__ m2,
                            const float* __restrict__ v2,
                            float* __restrict__ Wkc2, float* __restrict__ wdc2,
                            float* __restrict__ c3,
                            float* __restrict__ sm1, float* __restrict__ tm1,
                            float* __restrict__ sm2, float* __restrict__ tm2) {
  const int tid = blockIdx.x * blockDim.x + threadIdx.x;
  if (tid < 6144) {
    const int r = tid >> 7, c = tid & 127;
    const float s3 = g3[c] * rsqrtf(v3[c] + EPS_S);
    float acc = 0.f;
    for (int m = 0; m < 64; ++m) acc += Wk[r * 64 + m] * Wc[m * 128 + c];
    Wkc2[tid] = acc * s3;
  } else if (tid < 6272) {
    const int c = tid - 6144;
    const float s3 = g3[c] * rsqrtf(v3[c] + EPS_S);
    float acc = 0.f;
    for (int m = 0; m < 64; ++m) acc += Wd[m] * Wc[(64 + m) * 128 + c];
    wdc2[c] = acc * s3;
  } else if (tid < 6400) {
    const int c = tid - 6272;
    const float s3 = g3[c] * rsqrtf(v3[c] + EPS_S);
    float acc = bc[c];
    for (int m = 0; m < 64; ++m)
      acc += bk[m] * Wc[m * 128 + c] + bd[m] * Wc[(64 + m) * 128 + c];
    c3[c] = (acc - m3[c]) * s3 + be3[c];
  } else if (tid < 6912) {
    const int c = tid - 6400;
    const float s = g1[c] * rsqrtf(v1[c] + EPS_S);
    sm1[c] = s;
    tm1[c] = (bm1[c] - m1[c]) * s + be1[c];
  } else if (tid < 7424) {
    const int c = tid - 6912;
    const float s = g2[c] * rsqrtf(v2[c] + EPS_S);
    sm2[c] = s;
    tm2[c] = (bm2[c] - m2[c]) * s + be2[c];
  }
}

// ---------------------------------------------------------------------------
// Score (WMMA): per wave, 16 gathered rows (triples t=(q,j)) x K=48 against
// Wkc2 (48x128) in LDS. score[t] = max_c lrelu(acc + exp(-dist)*wdc2 + c3).
// Row-max reduced with __shfl_xor within each 16-lane half (rows 0-7 live in
// lanes 0-15, rows 8-15 in lanes 16-31 per the f32 C/D layout).
// ---------------------------------------------------------------------------
__global__ __launch_bounds__(256) void score_wmma_kernel(const float* __restrict__ x2,
                                                         const int* __restrict__ idx33,
                                                         const float* __restrict__ dist33,
                                                         const float* __restrict__ Wkc2,
                                                         const float* __restrict__ wdc2,
                                                         const float* __restrict__ c3,
                                                         float* __restrict__ score) {
  __shared__ float sW[48 * 128];      // 24.6 KB
  __shared__ float swd[128];
  __shared__ float sc3[128];
  __shared__ float sA[8][16][49];     // 49-pitch: gcd(49,64)=1 -> conflict-free
  __shared__ float se[8][16];
  const int tid  = threadIdx.x;
  const int wave = tid >> 5, lane = tid & 31;
  const int half = lane >> 4, l16 = lane & 15;

  for (int u = tid; u < 48 * 128; u += 256) sW[u] = Wkc2[u];
  if (tid < 128) { swd[tid] = wdc2[tid]; sc3[tid] = c3[tid]; }

  // gather: each lane stages half of one row (row = lane/2, 24 floats)
  const int tile     = blockIdx.x * 8 + wave;
  const int gid_base = tile * 16;                 // triple index base
  {
    const int row = lane >> 1;
    const int t   = gid_base + row;
    const int q   = t / KP1;
    const int b   = q / Nc;
    const int nb  = idx33[t];
    const float* src = x2 + ((size_t)b * Nc + nb) * 48;
    const int c0 = (lane & 1) * 24;
    #pragma unroll
    for (int m = 0; m < 24; ++m) sA[wave][row][c0 + m] = src[c0 + m];
    if ((lane & 1) == 0) se[wave][row] = __expf(-dist33[t]);
  }
  __syncthreads();

  float rowmax[8];
  #pragma unroll
  for (int r = 0; r < 8; ++r) rowmax[r] = -3.0e38f;

  for (int nt = 0; nt < 8; ++nt) {                // 8 N-tiles of 16 cols
    v8f acc = {};
    #pragma unroll
    for (int kk = 0; kk < 48; kk += 4) {
      v2f a, bf;
      const int ka = kk + 2 * half;
      a.x  = sA[wave][l16][ka + 0];
      a.y  = sA[wave][l16][ka + 1];
      bf.x = sW[(ka + 0) * 128 + nt * 16 + l16];
      bf.y = sW[(ka + 1) * 128 + nt * 16 + l16];
      acc = __builtin_amdgcn_wmma_f32_16x16x4_f32(false, a, false, bf,
                                                  (short)0, acc, false, false);
    }
    const int col = nt * 16 + l16;
    #pragma unroll
    for (int r = 0; r < 8; ++r) {
      const float e_r = se[wave][r + 8 * half];
      float v = acc[r] + e_r * swd[col] + sc3[col];
      rowmax[r] = fmaxf(rowmax[r], lrelu(v));
    }
  }
  // reduce over the 16 columns in each half
  #pragma unroll
  for (int r = 0; r < 8; ++r) {
    float v = rowmax[r];
    v = fmaxf(v, __shfl_xor(v, 1, 32));
    v = fmaxf(v, __shfl_xor(v, 2, 32));
    v = fmaxf(v, __shfl_xor(v, 4, 32));
    v = fmaxf(v, __shfl_xor(v, 8, 32));
    rowmax[r] = v;
  }
  if (l16 == 0) {
    #pragma unroll
    for (int r = 0; r < 8; ++r)
      score[(size_t)gid_base + r + 8 * half] = rowmax[r];
  }
}

// ---------------------------------------------------------------------------
// Top-10 of 33 scores per point (stable desc ties -> lower j), gather indices.
// ---------------------------------------------------------------------------
__global__ __launch_bounds__(256) void topk_kernel(const float* __restrict__ score,
                                                   const int* __restrict__ idx33,
                                                   int* __restrict__ toppts) {
  const int q = blockIdx.x * 256 + threadIdx.x;
  float sc[KP1];
  #pragma unroll
  for (int j = 0; j < KP1; ++j) sc[j] = score[(size_t)q * KP1 + j];
  unsigned long long used = 0ull;
  for (int t = 0; t < NKEY; ++t) {
    float best = -3.0e38f;
    int bj = 0;
    #pragma unroll
    for (int j = 0; j < KP1; ++j) {
      const bool free_slot = !((used >> j) & 1ull);
      if (free_slot && sc[j] > best) { best = sc[j]; bj = j; }
    }
    used |= (1ull << bj);
    toppts[(size_t)q * NKEY + t] = idx33[(size_t)q * KP1 + bj];
  }
}

__global__ __launch_bounds__(256) void gather_keyf_kernel(const float* __restrict__ x2,
                                                          const int* __restrict__ toppts,
                                                          float* __restrict__ keyf) {
  const int gid = blockIdx.x * 256 + threadIdx.x;   // [0, B*N*10)
  const int q = gid / NKEY, t = gid % NKEY;
  const int b = q / Nc;
  const int pt = toppts[gid];
  const float* src = x2 + ((size_t)b * Nc + pt) * 48;
  float* dst = keyf + (size_t)q * 480 + t * 48;
  #pragma unroll
  for (int c = 0; c < 48; ++c) dst[c] = src[c];
}

// ---------------------------------------------------------------------------
// WMMA fp32 GEMM: out = lrelu( (A@W)*scale[col] + shift[col] )
// A: MxK row-major, W: KxN row-major. M%32==0, N%64==0, K%32==0.
// Block = 256 threads = 8 waves; block tile 32(M) x 64(N); A staged into LDS
// with global_load_async_to_lds_b128 (one 16B transfer per thread per chunk,
// ASYNCcnt-tracked), pitch 36 floats -> rows 16B-aligned, conflict-free.
// ---------------------------------------------------------------------------
__global__ __launch_bounds__(256) void gemm_wmma_kernel(const float* __restrict__ A,
                                                        const float* __restrict__ W,
                                                        const float* __restrict__ scale,
                                                        const float* __restrict__ shift,
                                                        float* __restrict__ out,
                                                        int M, int N, int K) {
  __shared__ float sA[32][36];
  const int tid = threadIdx.x;
  const int wave = tid >> 5, lane = tid & 31;
  const int half = lane >> 4, l16 = lane & 15;
  const int waveM = wave >> 2, waveN = wave & 3;
  const int rowBase = blockIdx.y * 32;
  const int tileM = rowBase + waveM * 16;
  const int tileN = blockIdx.x * 64 + waveN * 16;

  // per-thread async-staging coordinates: 32 rows x 8 b128 columns
  const int srow = tid >> 3;            // 0..31
  const int scol = (tid & 7) * 4;       // 0,4,...,28
  const unsigned sA_dst = lds_off(&sA[srow][scol]);

  v8f acc = {};
  for (int k0 = 0; k0 < K; k0 += 32) {
    __syncthreads();   // protect sA against overwrite while still being read
    {
      const float* src = &A[(size_t)(rowBase + srow) * K + k0 + scol];
      asm volatile("global_load_async_to_lds_b128 %0, %1, off"
                   :: "v"(sA_dst), "v"(src) : "memory");
    }
    if (k0 + 32 < K)  // prefetch next A chunk -> global_prefetch_b8
      __builtin_prefetch(&A[(size_t)(rowBase + srow) * K + k0 + 32 + scol], 0, 1);
    asm volatile("s_wait_asynccnt 0" ::: "memory");
    __syncthreads();
    #pragma unroll
    for (int kk = 0; kk < 32; kk += 4) {
      v2f a, bf;
      a.x = sA[waveM * 16 + l16][kk + 2 * half + 0];
      a.y = sA[waveM * 16 + l16][kk + 2 * half + 1];
      const float* wp = W + (size_t)(k0 + kk + 2 * half) * N + tileN + l16;
      bf.x = wp[0];
      bf.y = wp[N];
      acc = __builtin_amdgcn_wmma_f32_16x16x4_f32(false, a, false, bf,
                                                  (short)0, acc, false, false);
    }
  }
  #pragma unroll
  for (int r = 0; r < 8; ++r) {
    const int row = tileM + half * 8 + r;
    const int col = tileN + l16;
    float v = acc[r] * scale[col] + shift[col];
    v = v > 0.f ? v : LRELU_S * v;
    out[(size_t)row * N + col] = v;
  }
}

// ---------------------------------------------------------------------------
// Host launcher
// ---------------------------------------------------------------------------
extern "C" void kernel_launch(void* const* d_in, const int* in_sizes, int n_in,
                              void* d_out, int out_size, void* d_ws, size_t ws_size,
                              hipStream_t stream) {
  const float* x   = (const float*)d_in[0];
  const float* W1  = (const float*)d_in[1];
  const float* b1  = (const float*)d_in[2];
  const float* W2  = (const float*)d_in[3];
  const float* b2  = (const float*)d_in[4];
  const float* Wd  = (const float*)d_in[5];
  const float* bd  = (const float*)d_in[6];
  const float* Wk  = (const float*)d_in[7];
  const float* bk  = (const float*)d_in[8];
  const float* Wc  = (const float*)d_in[9];
  const float* bc  = (const float*)d_in[10];
  const float* g3  = (const float*)d_in[11];
  const float* be3 = (const float*)d_in[12];
  const float* m3  = (const float*)d_in[13];
  const float* v3  = (const float*)d_in[14];
  const float* Wm1 = (const float*)d_in[15];
  const float* bm1 = (const float*)d_in[16];
  const float* g1  = (const float*)d_in[17];
  const float* be1 = (const float*)d_in[18];
  const float* m1  = (const float*)d_in[19];
  const float* v1  = (const float*)d_in[20];
  const float* Wm2 = (const float*)d_in[21];
  const float* bm2 = (const float*)d_in[22];
  const float* g2  = (const float*)d_in[23];
  const float* be2 = (const float*)d_in[24];
  const float* m2  = (const float*)d_in[25];
  const float* v2  = (const float*)d_in[26];
  // d_in[27]=k (32), d_in[28]=nkey (10): device scalars, fixed by setup_inputs.
  float* out = (float*)d_out;

  // workspace carve (256B aligned); total ~80 MB
  char* p = (char*)d_ws;
  auto carve = [&](size_t bytes) -> void* {
    void* r = (void*)p;
    p += (bytes + 255) & ~(size_t)255;
    return r;
  };
  float* dist33 = (float*)carve((size_t)BNc * KP1 * 4);
  int*   idx33  = (int*)  carve((size_t)BNc * KP1 * 4);
  int*   idx0   = (int*)  carve((size_t)BNc * KNBR * 4);
  int*   idx1   = (int*)  carve((size_t)BNc * KNBR * 4);
  float* x1     = (float*)carve((size_t)BNc * 16 * 4);
  float* x2     = (float*)carve((size_t)BNc * 48 * 4);
  float* score  = (float*)carve((size_t)BNc * KP1 * 4);
  int*   toppts = (int*)  carve((size_t)BNc * NKEY * 4);
  float* keyf   = (float*)carve((size_t)BNc * 480 * 4);
  float* h1     = (float*)carve((size_t)BNc * 512 * 4);
  float* Wkc2   = (float*)carve(48 * 128 * 4);
  float* wdc2   = (float*)carve(128 * 4);
  float* c3     = (float*)carve(128 * 4);
  float* sm1    = (float*)carve(512 * 4);
  float* tm1    = (float*)carve(512 * 4);
  float* sm2    = (float*)carve(512 * 4);
  float* tm2    = (float*)carve(512 * 4);
  (void)ws_size; (void)in_sizes; (void)n_in; (void)out_size;

  // 0) fold weights / BN params
  prep_kernel<<<29, 256, 0, stream>>>(Wk, bk, Wd, bd, Wc, bc, g3, be3, m3, v3,
                                      bm1, g1, be1, m1, v1, bm2, g2, be2, m2, v2,
                                      Wkc2, wdc2, c3, sm1, tm1, sm2, tm2);
  // 1) kNN passes
  knn_kernel<<<BNc / 128, 128, 0, stream>>>(x, 3, KP1, 0, dist33, idx33);  // incl. self, k+1
  knn_kernel<<<BNc / 128, 128, 0, stream>>>(x, 3, KNBR, 1, nullptr, idx0); // excl. self
  // 2) edge conv 1
  edgeconv1_kernel<<<BNc / 256, 256, 0, stream>>>(x, idx0, W1, b1, x1);
  // 3) kNN on x1
  knn_kernel<<<BNc / 128, 128, 0, stream>>>(x1, 16, KNBR, 1, nullptr, idx1);
  // 4) edge conv 2
  edgeconv2_kernel<<<BNc / 256, 256, 0, stream>>>(x1, idx1, W2, b2, x2);
  // 5) fused score on WMMA (gather + Wk + dist-feat + Wc + bn3 + lrelu + max)
  score_wmma_kernel<<<(BNc * KP1) / (16 * 8), 256, 0, stream>>>(x2, idx33, dist33,
                                                                Wkc2, wdc2, c3, score);
  // 6) top-10 key points + gather key features
  topk_kernel<<<BNc / 256, 256, 0, stream>>>(score, idx33, toppts);
  gather_keyf_kernel<<<(BNc * NKEY) / 256, 256, 0, stream>>>(x2, toppts, keyf);
  // 7) MLP head: two WMMA fp32 GEMMs with fused bias/BN/lrelu epilogues
  {
    dim3 g1d(512 / 64, BNc / 32);
    gemm_wmma_kernel<<<g1d, 256, 0, stream>>>(keyf, Wm1, sm1, tm1, h1, BNc, 512, 480);
    gemm_wmma_kernel<<<g1d, 256, 0, stream>>>(h1, Wm2, sm2, tm2, out, BNc, 512, 512);
  }
}